// GNNModel_21105469292957
// MI455X (gfx1250) — compile-verified
//
#include <hip/hip_runtime.h>
#include <hip/hip_bf16.h>

typedef __attribute__((ext_vector_type(2))) float v2f;
typedef __attribute__((ext_vector_type(4))) float v4f;
typedef __attribute__((ext_vector_type(8))) float v8f;

#define D 64
#define EPS_C 0.1f
#define GAMMA_C 0.1f

// Guaranteed-native CDNA5 FP32 atomic add (no return -> STOREcnt-tracked,
// resolves in L2; ordering across kernels comes from launch boundaries).
__device__ __forceinline__ void atomic_add_f32_native(float* p, float v) {
  asm volatile("global_atomic_add_f32 %0, %1, off" : : "v"(p), "v"(v) : "memory");
}

// ---- weight prep: transposed phiW and aW = (W - W^T - gamma*I), per layer ----
// Y = X @ M^T  ==>  B[k][n] = M[n][k]  (row-major [k*64+n] so GEMM B-loads coalesce)
__global__ void prep_weights(const float* __restrict__ W1, const float* __restrict__ phiW1,
                             const float* __restrict__ W2, const float* __restrict__ phiW2,
                             float* __restrict__ wts) {
  int t = blockIdx.x * blockDim.x + threadIdx.x;   // 0..4095
  int k = t >> 6, n = t & 63;
  float diag = (k == n) ? GAMMA_C : 0.0f;
  wts[t]         = phiW1[n * D + k];
  wts[4096 + t]  = W1[n * D + k] - W1[k * D + n] - diag;   // aW1^T as B[k][n]
  wts[8192 + t]  = phiW2[n * D + k];
  wts[12288 + t] = W2[n * D + k] - W2[k * D + n] - diag;
}

// ---- degree / normalization ----
__global__ void init_deg(float* __restrict__ deg, int n) {
  int i = blockIdx.x * blockDim.x + threadIdx.x;
  if (i < n) deg[i] = 1.0f;                         // self-loop contribution
}

__global__ void deg_scatter(const int* __restrict__ col, float* __restrict__ deg, int e) {
  int i = blockIdx.x * blockDim.x + threadIdx.x;
  if (i < e) atomic_add_f32_native(&deg[col[i]], 1.0f);
}

__global__ void deg_to_dinv(float* __restrict__ deg, int n) {
  int i = blockIdx.x * blockDim.x + threadIdx.x;
  if (i < n) deg[i] = rsqrtf(deg[i]);               // deg >= 1 always (self-loops)
}

// ---- WMMA GEMM: Y[M x 64] = X[M x 64] @ B[64 x 64] (+ bias[n]); M % 16 == 0 ----
// block = 128 threads = 4 waves; wave w computes columns [16w, 16w+16) of one 16-row tile.
__global__ __launch_bounds__(128) void gemm_wmma_64(const float* __restrict__ X,
                                                    const float* __restrict__ B,
                                                    const float* __restrict__ bias,
                                                    float* __restrict__ Y) {
  __shared__ float Bs[D * D];                       // 16 KB staged weights
  for (int i = threadIdx.x; i < D * D; i += 128) Bs[i] = B[i];
  __syncthreads();

  const int lane = threadIdx.x & 31;
  const int wv   = threadIdx.x >> 5;                // n-tile 0..3
  const int l    = lane & 15;
  const int half = lane >> 4;                       // 0 or 1
  const int m0   = blockIdx.x * 16;
  const int n    = wv * 16 + l;                     // output column for this lane

  float c0 = bias ? bias[n] : 0.0f;
  v8f acc;
#pragma unroll
  for (int r = 0; r < 8; ++r) acc[r] = c0;

  // A layout (16x4 f32): lane(0-15)->M=lane, K = k0 + 2*half + {0,1}
  const float* arow = X + (size_t)(m0 + l) * D + 2 * half;
#pragma unroll
  for (int k0 = 0; k0 < D; k0 += 4) {
    v2f a = *(const v2f*)(arow + k0);
    v2f b;
    b.x = Bs[(k0 + 2 * half + 0) * D + n];
    b.y = Bs[(k0 + 2 * half + 1) * D + n];
    acc = __builtin_amdgcn_wmma_f32_16x16x4_f32(false, a, false, b, (short)0, acc,
                                                false, false);
  }

  // C/D layout: VGPR r holds M = r + 8*half, N = n
#pragma unroll
  for (int r = 0; r < 8; ++r)
    Y[(size_t)(m0 + r + 8 * half) * D + n] = acc[r];
}

// ---- aggregation: self-loop init (vectorized b128 streaming) ----
__global__ void init_agg4(const v4f* __restrict__ xw, const float* __restrict__ dinv,
                          v4f* __restrict__ agg, int total4) {
  int i = blockIdx.x * blockDim.x + threadIdx.x;
  if (i < total4) {
    float dv = dinv[i >> 4];                        // 16 v4f per node row
    float s = dv * dv;
    v4f a = xw[i];
    a.x *= s; a.y *= s; a.z *= s; a.w *= s;
    agg[i] = a;
  }
}

// ---- edge scatter: 64 threads (2 waves) per edge; coalesced gather + L2 atomics ----
__global__ void edge_scatter(const int* __restrict__ row, const int* __restrict__ col,
                             const float* __restrict__ dinv, const float* __restrict__ xw,
                             float* __restrict__ agg, int e) {
  int t = blockIdx.x * blockDim.x + threadIdx.x;
  int ed = t >> 6, d = t & 63;
  if (ed < e) {
    int r = row[ed], c = col[ed];
    float w = dinv[r] * dinv[c];
    atomic_add_f32_native(&agg[(size_t)c * D + d], xw[(size_t)r * D + d] * w);
  }
}

// ---- x_next = relu(x + eps * tanh(z + agg)), vectorized b128 ----
__global__ void update_x4(const v4f* __restrict__ x, const v4f* __restrict__ z,
                          const v4f* __restrict__ agg, v4f* __restrict__ xo, int total4) {
  int i = blockIdx.x * blockDim.x + threadIdx.x;
  if (i < total4) {
    v4f zz = z[i], aa = agg[i], xx = x[i], o;
    o.x = fmaxf(fmaf(EPS_C, tanhf(zz.x + aa.x), xx.x), 0.0f);
    o.y = fmaxf(fmaf(EPS_C, tanhf(zz.y + aa.y), xx.y), 0.0f);
    o.z = fmaxf(fmaf(EPS_C, tanhf(zz.z + aa.z), xx.z), 0.0f);
    o.w = fmaxf(fmaf(EPS_C, tanhf(zz.w + aa.w), xx.w), 0.0f);
    xo[i] = o;
  }
}

// ---- out[i] = dot(x[i,:], outW) + outb ----
__global__ void out_proj(const float* __restrict__ X, const float* __restrict__ outW,
                         const float* __restrict__ outb, float* __restrict__ out, int n) {
  int i = blockIdx.x * blockDim.x + threadIdx.x;
  if (i >= n) return;
  const v4f* xr = (const v4f*)(X + (size_t)i * D);
  const v4f* wr = (const v4f*)outW;
  float s = 0.0f;
#pragma unroll
  for (int j = 0; j < D / 4; ++j) {
    v4f a = xr[j], w = wr[j];
    s += a.x * w.x + a.y * w.y + a.z * w.z + a.w * w.w;
  }
  out[i] = s + outb[0];
}

extern "C" void kernel_launch(void* const* d_in, const int* in_sizes, int n_in,
                              void* d_out, int out_size, void* d_ws, size_t ws_size,
                              hipStream_t stream) {
  const float* x     = (const float*)d_in[0];
  const float* W1    = (const float*)d_in[1];
  const float* phiW1 = (const float*)d_in[2];
  const float* b1    = (const float*)d_in[3];
  const float* W2    = (const float*)d_in[4];
  const float* phiW2 = (const float*)d_in[5];
  const float* b2    = (const float*)d_in[6];
  const float* outW  = (const float*)d_in[7];
  const float* outb  = (const float*)d_in[8];
  const int*   eidx  = (const int*)d_in[9];
  float* out = (float*)d_out;

  const int N = in_sizes[0] / D;     // 100000 (divisible by 16)
  const int E = in_sizes[9] / 2;     // 1200000
  const int* row = eidx;
  const int* col = eidx + E;

  // workspace layout (floats): dinv[N] | wts[4*4096] | xw/z[N*64] | agg[N*64] | xbuf[N*64]
  float* dinv = (float*)d_ws;
  float* wts  = dinv + N;
  float* xw   = wts + 4 * 4096;      // z aliases xw (z GEMM runs after scatter finished)
  float* agg  = xw + (size_t)N * D;
  float* xbuf = agg + (size_t)N * D;

  const int nd  = N * D;
  const int nd4 = nd / 4;

  prep_weights<<<16, 256, 0, stream>>>(W1, phiW1, W2, phiW2, wts);
  init_deg<<<(N + 255) / 256, 256, 0, stream>>>(dinv, N);
  deg_scatter<<<(E + 255) / 256, 256, 0, stream>>>(col, dinv, E);
  deg_to_dinv<<<(N + 255) / 256, 256, 0, stream>>>(dinv, N);

  const float* xcur = x;
  const float* biases[2] = {b1, b2};
  for (int l = 0; l < 2; ++l) {
    const float* phiWt = wts + l * 8192;
    const float* aWt   = wts + l * 8192 + 4096;
    gemm_wmma_64<<<N / 16, 128, 0, stream>>>(xcur, phiWt, nullptr, xw);
    init_agg4<<<(nd4 + 255) / 256, 256, 0, stream>>>((const v4f*)xw, dinv, (v4f*)agg, nd4);
    edge_scatter<<<(int)(((size_t)E * D + 255) / 256), 256, 0, stream>>>(row, col, dinv, xw, agg, E);
    gemm_wmma_64<<<N / 16, 128, 0, stream>>>(xcur, aWt, biases[l], xw);   // z = x@aW^T + b
    update_x4<<<(nd4 + 255) / 256, 256, 0, stream>>>((const v4f*)xcur, (const v4f*)xw,
                                                     (const v4f*)agg, (v4f*)xbuf, nd4);
    xcur = xbuf;                     // layer-2 update is same-index in-place: safe
  }
  out_proj<<<(N + 255) / 256, 256, 0, stream>>>(xcur, outW, outb, out, N);
}